// Entorpy_69458211111371
// MI455X (gfx1250) — compile-verified
//
#include <hip/hip_runtime.h>
#include <hip/hip_bf16.h>
#include <math.h>
#include <stdint.h>

#define NCLS 100
#define ROWS_PER_BLOCK 128
#define THREADS 128
#define ROW_BYTES (NCLS * 4)                         // 400 B per row
#define TILE_BYTES (ROWS_PER_BLOCK * ROW_BYTES)     // 51200 B LDS tile
#define CHUNKS_PER_ROW (ROW_BYTES / 16)              // 25 x b128 per row

typedef int v4i __attribute__((ext_vector_type(4)));
typedef __attribute__((address_space(1))) v4i* glob_v4i_ptr;
typedef __attribute__((address_space(3))) v4i* lds_v4i_ptr;

// ---------------------------------------------------------------- zero hist
__global__ void zero_kernel(unsigned* __restrict__ g_hist, float* __restrict__ out) {
    int t = threadIdx.x;
    if (t < NCLS) g_hist[t] = 0u;
    if (t == 0) out[0] = 0.0f;
}

// ---------------------------------------------------------------- histogram
__global__ __launch_bounds__(THREADS)
void hist_kernel(const float* __restrict__ logits, unsigned* __restrict__ g_hist,
                 int rows) {
    __shared__ float tile[ROWS_PER_BLOCK * NCLS];   // 51200 B
    __shared__ unsigned hist[NCLS];

    const int t = threadIdx.x;
    if (t < NCLS) hist[t] = 0u;

    const int row0 = blockIdx.x * ROWS_PER_BLOCK;
    int local_rows = rows - row0;
    if (local_rows > ROWS_PER_BLOCK) local_rows = ROWS_PER_BLOCK;
    if (local_rows <= 0) return;

    const char* gbase = (const char*)logits + (size_t)row0 * ROW_BYTES;

#if defined(__HIP_DEVICE_COMPILE__) && __has_builtin(__builtin_amdgcn_global_load_async_to_lds_b128)
    {
        __attribute__((address_space(3))) char* ldsb =
            (__attribute__((address_space(3))) char*)tile;
        if (local_rows == ROWS_PER_BLOCK) {
            // Fast path: full tile, 25 back-to-back async DMA issues per lane
            // (ASYNCcnt is 6 bits -> 25 outstanding is fine), fully contiguous
            // global addresses across the wave (512 B per wave per issue).
            #pragma unroll
            for (int i = 0; i < CHUNKS_PER_ROW; ++i) {
                const int k = t + i * THREADS;
                __builtin_amdgcn_global_load_async_to_lds_b128(
                    (glob_v4i_ptr)(gbase + (size_t)k * 16),
                    (lds_v4i_ptr)(ldsb + k * 16),
                    0, 0);
            }
        } else {
            const int nchunks = local_rows * CHUNKS_PER_ROW;
            for (int k = t; k < nchunks; k += THREADS) {
                __builtin_amdgcn_global_load_async_to_lds_b128(
                    (glob_v4i_ptr)(gbase + (size_t)k * 16),
                    (lds_v4i_ptr)(ldsb + k * 16),
                    0, 0);
            }
        }
#if __has_builtin(__builtin_amdgcn_s_wait_asynccnt)
        __builtin_amdgcn_s_wait_asynccnt(0);
#else
        asm volatile("s_wait_asynccnt 0" ::: "memory");
#endif
    }
#else
    // Fallback (host pass / builtin missing): plain vector copy into LDS.
    {
        const int nchunks = local_rows * CHUNKS_PER_ROW;
        const float4* gsrc = (const float4*)gbase;
        float4* ldst = (float4*)tile;
        for (int k = t; k < nchunks; k += THREADS) ldst[k] = gsrc[k];
    }
#endif
    __syncthreads();

    // Each thread argmax-scans its own row from LDS (stride 100 dwords -> 2-way bank conflict max)
    if (t < local_rows) {
        const float4* row = (const float4*)(tile + t * NCLS);
        float best = -INFINITY;
        int bidx = 0;
        #pragma unroll
        for (int i = 0; i < NCLS / 4; ++i) {
            float4 v = row[i];
            int c = i * 4;
            if (v.x > best) { best = v.x; bidx = c;     }   // strict > keeps first max
            if (v.y > best) { best = v.y; bidx = c + 1; }
            if (v.z > best) { best = v.z; bidx = c + 2; }
            if (v.w > best) { best = v.w; bidx = c + 3; }
        }
        atomicAdd(&hist[bidx], 1u);                       // ds_add_u32
    }
    __syncthreads();

    if (t < NCLS) {
        unsigned h = hist[t];
        if (h) atomicAdd(&g_hist[t], h);                  // global_atomic_add_u32
    }
}

// ---------------------------------------------------------------- entropy
__global__ __launch_bounds__(128)
void entropy_kernel(const unsigned* __restrict__ g_hist, float* __restrict__ out,
                    float inv_b) {
    __shared__ float partial[128];
    int t = threadIdx.x;
    float term = 0.0f;
    if (t < NCLS) {
        unsigned h = g_hist[t];
        if (h > 0u) {
            float p = (float)h * inv_b;
            term = p * log2f(p);
        }
    }
    partial[t] = term;
    __syncthreads();
    #pragma unroll
    for (int s = 64; s > 0; s >>= 1) {
        if (t < s) partial[t] += partial[t + s];
        __syncthreads();
    }
    if (t == 0) out[0] = -partial[0];
}

// ---------------------------------------------------------------- launch
extern "C" void kernel_launch(void* const* d_in, const int* in_sizes, int n_in,
                              void* d_out, int out_size, void* d_ws, size_t ws_size,
                              hipStream_t stream) {
    const float* logits = (const float*)d_in[0];
    unsigned* g_hist = (unsigned*)d_ws;          // 100 x u32 scratch
    float* out = (float*)d_out;

    const int rows = in_sizes[0] / NCLS;         // B = 4,194,304
    const int nblocks = (rows + ROWS_PER_BLOCK - 1) / ROWS_PER_BLOCK;

    zero_kernel<<<1, 128, 0, stream>>>(g_hist, out);
    hist_kernel<<<nblocks, THREADS, 0, stream>>>(logits, g_hist, rows);
    entropy_kernel<<<1, 128, 0, stream>>>(g_hist, out, 1.0f / (float)rows);
}